// ConceptGaussians_45973329936461
// MI455X (gfx1250) — compile-verified
//
#include <hip/hip_runtime.h>
#include <hip/hip_bf16.h>

// ConceptGaussians gather: out_means[b,d] = mean[d, labels[b,d]],
//                          out_logv [b,d] = log_var[d, labels[b,d]]
// B = 2^21, D = 8, C = 64. Pure bandwidth problem (~192 MB @ 23.3 TB/s ~ 8 us).
// CDNA5 paths: async global->LDS staging of the 2x2KB tables (ASYNCcnt),
// paired DS gathers, and non-temporal hints on the single-touch streams.

#define NDOM    8
#define NCON    64
#define THREADS 256
#define ITEMS   4

// Native clang vectors (required by __builtin_nontemporal_*; HIP's int4/float4
// are structs and are rejected).
typedef int   v4i __attribute__((ext_vector_type(4)));
typedef float v4f __attribute__((ext_vector_type(4)));

__global__ __launch_bounds__(THREADS) void concept_gauss_kernel(
    const int*   __restrict__ labels,
    const float* __restrict__ mean,
    const float* __restrict__ log_var,
    float*       __restrict__ out,
    int B)
{
    // [0,512) = mean (8x64), [512,1024) = log_var (8x64)
    __shared__ float s_tab[2 * NDOM * NCON];

    // ---- Stage both tables into LDS with gfx1250 async global->LDS DMA ----
    // 256 lanes * 8 bytes = 2048 bytes = one full table per instruction.
    {
        const unsigned t = threadIdx.x;
        unsigned lds_m = (unsigned)(uintptr_t)(&s_tab[t * 2]);
        unsigned lds_v = (unsigned)(uintptr_t)(&s_tab[NDOM * NCON + t * 2]);
        unsigned long long ga_m = (unsigned long long)(uintptr_t)(mean    + t * 2);
        unsigned long long ga_v = (unsigned long long)(uintptr_t)(log_var + t * 2);
        asm volatile("global_load_async_to_lds_b64 %0, %1, off"
                     :: "v"(lds_m), "v"(ga_m) : "memory");
        asm volatile("global_load_async_to_lds_b64 %0, %1, off"
                     :: "v"(lds_v), "v"(ga_v) : "memory");
        // Wait for this wave's async copies to land in LDS (ASYNCcnt).
        asm volatile("s_wait_asynccnt 0x0" ::: "memory");
    }
    __syncthreads();

    const size_t blockBase = (size_t)blockIdx.x * (THREADS * ITEMS);
    float* __restrict__ outV = out + (size_t)B * NDOM;   // log_var half of d_out

#pragma unroll
    for (int it = 0; it < ITEMS; ++it) {
        const size_t b = blockBase + (size_t)it * THREADS + threadIdx.x;
        if (b >= (size_t)B) continue;

        // 32B coalesced label load, non-temporal (single touch)
        const v4i* __restrict__ lab = (const v4i*)(labels + b * NDOM);
        const v4i l0 = __builtin_nontemporal_load(&lab[0]);
        const v4i l1 = __builtin_nontemporal_load(&lab[1]);

        // 16 per-lane LDS gathers (bank = concept index, 64 banks);
        // compiler pairs the mean/log_var gathers into 2-address DS loads.
        v4f m0, m1, v0, v1;
        m0.x = s_tab[0 * NCON + (l0.x & 63)];
        m0.y = s_tab[1 * NCON + (l0.y & 63)];
        m0.z = s_tab[2 * NCON + (l0.z & 63)];
        m0.w = s_tab[3 * NCON + (l0.w & 63)];
        m1.x = s_tab[4 * NCON + (l1.x & 63)];
        m1.y = s_tab[5 * NCON + (l1.y & 63)];
        m1.z = s_tab[6 * NCON + (l1.z & 63)];
        m1.w = s_tab[7 * NCON + (l1.w & 63)];

        v0.x = s_tab[NDOM * NCON + 0 * NCON + (l0.x & 63)];
        v0.y = s_tab[NDOM * NCON + 1 * NCON + (l0.y & 63)];
        v0.z = s_tab[NDOM * NCON + 2 * NCON + (l0.z & 63)];
        v0.w = s_tab[NDOM * NCON + 3 * NCON + (l0.w & 63)];
        v1.x = s_tab[NDOM * NCON + 4 * NCON + (l1.x & 63)];
        v1.y = s_tab[NDOM * NCON + 5 * NCON + (l1.y & 63)];
        v1.z = s_tab[NDOM * NCON + 6 * NCON + (l1.z & 63)];
        v1.w = s_tab[NDOM * NCON + 7 * NCON + (l1.w & 63)];

        // 64B coalesced non-temporal stores (four b128s per lane, no reuse)
        v4f* __restrict__ om = (v4f*)(out  + b * NDOM);
        __builtin_nontemporal_store(m0, &om[0]);
        __builtin_nontemporal_store(m1, &om[1]);
        v4f* __restrict__ ov = (v4f*)(outV + b * NDOM);
        __builtin_nontemporal_store(v0, &ov[0]);
        __builtin_nontemporal_store(v1, &ov[1]);
    }
}

extern "C" void kernel_launch(void* const* d_in, const int* in_sizes, int n_in,
                              void* d_out, int out_size, void* d_ws, size_t ws_size,
                              hipStream_t stream) {
    const int*   labels  = (const int*)d_in[0];    // [B, 8] int32
    const float* mean    = (const float*)d_in[1];  // [8, 64] f32
    const float* log_var = (const float*)d_in[2];  // [8, 64] f32
    float*       out     = (float*)d_out;          // [B,8] means ++ [B,8] log_vars

    const int B = in_sizes[0] / NDOM;
    const int perBlock = THREADS * ITEMS;
    const int blocks = (B + perBlock - 1) / perBlock;

    hipLaunchKernelGGL(concept_gauss_kernel, dim3(blocks), dim3(THREADS), 0, stream,
                       labels, mean, log_var, out, B);
}